// DomainAdversarialModel_1967095021743
// MI455X (gfx1250) — compile-verified
//
#include <hip/hip_runtime.h>
#include <hip/hip_bf16.h>

// ---------------------------------------------------------------------------
// CDNA5 / gfx1250 implementation of the DomainAdversarial GNN reference.
// Key transform: x[src] @ W == (x @ W)[src]  -> dense WMMA GEMMs + edge scatter.
// A-panel staging uses the gfx1250 async global->LDS path (ASYNCcnt).
// ---------------------------------------------------------------------------

typedef __attribute__((ext_vector_type(2))) float v2f;
typedef __attribute__((ext_vector_type(8))) float v8f;
typedef __attribute__((ext_vector_type(4))) int   v4i;

typedef __attribute__((address_space(1))) v4i* gv4i_ptr;   // global int4*
typedef __attribute__((address_space(3))) v4i* lv4i_ptr;   // LDS int4*

#define D 128
#define NGRAPHS 64
#define HID 64      // D/2
#define NPAT 50

__device__ __forceinline__ float elu_f(float x) {
    return x > 0.0f ? x : (__expf(x) - 1.0f);
}

__device__ __forceinline__ float wave_reduce_sum(float s) {
    #pragma unroll
    for (int off = 16; off >= 1; off >>= 1)
        s += __shfl_xor(s, off, 32);
    return s;
}

__device__ __forceinline__ void async_copy_b128(const float* gp, float* lp) {
#if __has_builtin(__builtin_amdgcn_global_load_async_to_lds_b128)
    __builtin_amdgcn_global_load_async_to_lds_b128(
        (gv4i_ptr)const_cast<float*>(gp), (lv4i_ptr)lp, 0, 0);
#else
    asm volatile("global_load_async_to_lds_b128 %0, %1, off"
                 :: "v"((lv4i_ptr)lp), "v"((gv4i_ptr)const_cast<float*>(gp))
                 : "memory");
#endif
}

__device__ __forceinline__ void wait_asynccnt0() {
#if __has_builtin(__builtin_amdgcn_s_wait_asynccnt)
    __builtin_amdgcn_s_wait_asynccnt(0);
#else
    asm volatile("s_wait_asynccnt 0x0" ::: "memory");
#endif
}

// ---------------------------------------------------------------------------
// Fused dual GEMM:  H = X @ Wr + b   and   Y = X @ Wl
// X: [nrows x 128] row-major, Wr/Wl: [128 x 128] row-major.
// One block = 256 threads = 8 waves; block owns a 16-row panel of X, staged
// into LDS with GLOBAL_LOAD_ASYNC_TO_LDS_B128; wave w computes the 16x16
// output tile at columns [16w, 16w+16) for BOTH weight matrices using
// V_WMMA_F32_16X16X4_F32 (fp32 in, fp32 acc).
// ---------------------------------------------------------------------------
#define AS_STRIDE 132   // 128 + 4 pad -> spreads LDS banks (132 % 64 == 4)

__global__ void __launch_bounds__(256)
gemm2_wmma_kernel(const float* __restrict__ X,
                  const float* __restrict__ Wr,
                  const float* __restrict__ Wl,
                  const float* __restrict__ bias,
                  float* __restrict__ H,
                  float* __restrict__ Y) {
    __shared__ float As[16 * AS_STRIDE];

    const int tid = threadIdx.x;
    const int M0  = blockIdx.x * 16;

    // Stage 16x128 A panel into LDS: 512 x 16-byte async chunks.
    // Row stride is 132 floats (pad), chunks of 4 floats never straddle pad.
    {
        const float* gsrc = X + (size_t)M0 * D;
        #pragma unroll
        for (int c = 0; c < 2; ++c) {
            int chunk = tid + c * 256;         // 0..511
            int row   = chunk >> 5;            // 0..15
            int col4  = (chunk & 31) * 4;      // 0,4,...,124
            async_copy_b128(gsrc + row * D + col4,
                            &As[row * AS_STRIDE + col4]);
        }
        wait_asynccnt0();
        __syncthreads();
    }

    const int wave = tid >> 5;            // 0..7 -> column tile
    const int lane = tid & 31;
    const int lm   = lane & 15;           // M (for A) / N (for B,C,D)
    const int hi   = lane >> 4;           // 0: K={0,1}, 1: K={2,3}
    const int N0   = wave * 16;

    v8f acc_r = {0.f,0.f,0.f,0.f,0.f,0.f,0.f,0.f};
    v8f acc_l = {0.f,0.f,0.f,0.f,0.f,0.f,0.f,0.f};

    #pragma unroll 4
    for (int k0 = 0; k0 < D; k0 += 4) {
        const int ka = k0 + 2 * hi;       // this half-wave's K base
        // A tile 16x4: lane m=lm, VGPR j holds K = ka + j
        v2f a;
        a.x = As[lm * AS_STRIDE + ka];
        a.y = As[lm * AS_STRIDE + ka + 1];
        // B tile 4x16: lane n=lm, VGPR j holds K = ka + j
        v2f br, bl;
        br.x = Wr[(ka    ) * D + N0 + lm];
        br.y = Wr[(ka + 1) * D + N0 + lm];
        bl.x = Wl[(ka    ) * D + N0 + lm];
        bl.y = Wl[(ka + 1) * D + N0 + lm];

        acc_r = __builtin_amdgcn_wmma_f32_16x16x4_f32(false, a, false, br,
                                                      (short)0, acc_r, false, false);
        acc_l = __builtin_amdgcn_wmma_f32_16x16x4_f32(false, a, false, bl,
                                                      (short)0, acc_l, false, false);
    }

    // C/D layout: VGPR i -> row M0 + i + 8*hi, col N0 + lm
    const int   col = N0 + lm;
    const float bv  = bias[col];
    #pragma unroll
    for (int i = 0; i < 8; ++i) {
        int row = M0 + i + 8 * hi;
        H[row * D + col] = acc_r[i] + bv;
        Y[row * D + col] = acc_l[i];
    }
}

// ---------------------------------------------------------------------------
// Edge scatter: H[dst] += Y[src]  (one wave per edge, float4 per lane,
// global_atomic_add_f32 for accumulation).
// ---------------------------------------------------------------------------
__global__ void __launch_bounds__(256)
edge_scatter_kernel(const float* __restrict__ Y,
                    const int* __restrict__ src,
                    const int* __restrict__ dst,
                    float* __restrict__ H, int E) {
    int gid  = blockIdx.x * blockDim.x + threadIdx.x;
    int e    = gid >> 5;
    int lane = gid & 31;
    if (e >= E) return;
    int s = src[e];
    int d = dst[e];
    const float4 v = *(const float4*)(Y + (long)s * D + lane * 4);
    float* p = H + (long)d * D + lane * 4;
    atomicAdd(p + 0, v.x);
    atomicAdd(p + 1, v.y);
    atomicAdd(p + 2, v.z);
    atomicAdd(p + 3, v.w);
}

// ---------------------------------------------------------------------------
// LayerNorm + ELU + residual (in place on Xc): one wave per node row.
// ---------------------------------------------------------------------------
__global__ void __launch_bounds__(256)
ln_elu_res_kernel(const float* __restrict__ H,
                  float* __restrict__ Xc,
                  const float* __restrict__ gamma,
                  const float* __restrict__ beta,
                  int n) {
    int gid  = blockIdx.x * blockDim.x + threadIdx.x;
    int node = gid >> 5;
    int lane = gid & 31;
    if (node >= n) return;

    const float4 h = *(const float4*)(H + (long)node * D + lane * 4);
    float s  = h.x + h.y + h.z + h.w;
    float ss = h.x * h.x + h.y * h.y + h.z * h.z + h.w * h.w;
    s  = wave_reduce_sum(s);
    ss = wave_reduce_sum(ss);
    const float mean = s * (1.0f / D);
    const float var  = ss * (1.0f / D) - mean * mean;
    const float rstd = __frsqrt_rn(var + 1e-5f);

    const float4 g = *(const float4*)(gamma + lane * 4);
    const float4 b = *(const float4*)(beta  + lane * 4);
    float4 xi = *(float4*)(Xc + (long)node * D + lane * 4);

    float4 o;
    o.x = elu_f((h.x - mean) * rstd * g.x + b.x) + xi.x;
    o.y = elu_f((h.y - mean) * rstd * g.y + b.y) + xi.y;
    o.z = elu_f((h.z - mean) * rstd * g.z + b.z) + xi.z;
    o.w = elu_f((h.w - mean) * rstd * g.w + b.w) + xi.w;
    *(float4*)(Xc + (long)node * D + lane * 4) = o;
}

// ---------------------------------------------------------------------------
// Boundary head: out[i] = dot(x[i], bW) + bb   (one wave per node)
// ---------------------------------------------------------------------------
__global__ void __launch_bounds__(256)
boundary_kernel(const float* __restrict__ Xc,
                const float* __restrict__ bW,
                const float* __restrict__ bb,
                float* __restrict__ out, int n) {
    int gid  = blockIdx.x * blockDim.x + threadIdx.x;
    int node = gid >> 5;
    int lane = gid & 31;
    if (node >= n) return;
    const float4 x = *(const float4*)(Xc + (long)node * D + lane * 4);
    const float4 w = *(const float4*)(bW + lane * 4);
    float s = x.x * w.x + x.y * w.y + x.z * w.z + x.w * w.w;
    s = wave_reduce_sum(s);
    if (lane == 0) out[node] = s + bb[0];
}

// ---------------------------------------------------------------------------
// Pooling: pooled[batch[i]] += x[i]; counts[batch[i]] += 1
// ---------------------------------------------------------------------------
__global__ void zero_kernel(float* __restrict__ p, int n) {
    int i = blockIdx.x * blockDim.x + threadIdx.x;
    if (i < n) p[i] = 0.0f;
}

__global__ void __launch_bounds__(256)
pool_kernel(const float* __restrict__ Xc,
            const int* __restrict__ batch,
            float* __restrict__ pooled,
            float* __restrict__ counts, int n) {
    int gid  = blockIdx.x * blockDim.x + threadIdx.x;
    int node = gid >> 5;
    int lane = gid & 31;
    if (node >= n) return;
    int g = batch[node];
    const float4 v = *(const float4*)(Xc + (long)node * D + lane * 4);
    float* p = pooled + (long)g * D + lane * 4;
    atomicAdd(p + 0, v.x);
    atomicAdd(p + 1, v.y);
    atomicAdd(p + 2, v.z);
    atomicAdd(p + 3, v.w);
    if (lane == 0) atomicAdd(&counts[g], 1.0f);
}

// ---------------------------------------------------------------------------
// Domain head: mean -> GRL(identity fwd) -> elu(r@W1+b1) -> @W2+b2
// Single block; tiny problem (64x128 -> 64x64 -> 64x50).
// ---------------------------------------------------------------------------
__global__ void __launch_bounds__(256)
domain_head_kernel(const float* __restrict__ pooled,
                   const float* __restrict__ counts,
                   const float* __restrict__ W1,   // [128 x 64] row-major
                   const float* __restrict__ b1,   // [64]
                   const float* __restrict__ W2,   // [64 x 50] row-major
                   const float* __restrict__ b2,   // [50]
                   float* __restrict__ out) {      // [64 x 50]
    __shared__ float pm[NGRAPHS * D];      // 32 KB
    __shared__ float h1[NGRAPHS * HID];    // 16 KB
    const int tid = threadIdx.x;

    for (int i = tid; i < NGRAPHS * D; i += 256) {
        int g = i >> 7;
        float c = counts[g];
        c = c < 1.0f ? 1.0f : c;
        pm[i] = pooled[i] / c;
    }
    __syncthreads();

    for (int i = tid; i < NGRAPHS * HID; i += 256) {
        int g = i >> 6;
        int j = i & 63;
        float s = b1[j];
        #pragma unroll 8
        for (int k = 0; k < D; ++k)
            s += pm[g * D + k] * W1[k * HID + j];
        h1[i] = elu_f(s);
    }
    __syncthreads();

    for (int i = tid; i < NGRAPHS * NPAT; i += 256) {
        int g = i / NPAT;
        int p = i - g * NPAT;
        float s = b2[p];
        #pragma unroll 8
        for (int k = 0; k < HID; ++k)
            s += h1[g * HID + k] * W2[k * NPAT + p];
        out[i] = s;
    }
}

// ---------------------------------------------------------------------------
// Utility: copy kernel (x -> workspace copy we can mutate in place)
// ---------------------------------------------------------------------------
__global__ void copy_kernel(const float* __restrict__ src,
                            float* __restrict__ dst, int n) {
    int i = blockIdx.x * blockDim.x + threadIdx.x;
    if (i < n) dst[i] = src[i];
}

// ---------------------------------------------------------------------------
// Host launcher
// ---------------------------------------------------------------------------
extern "C" void kernel_launch(void* const* d_in, const int* in_sizes, int n_in,
                              void* d_out, int out_size, void* d_ws, size_t ws_size,
                              hipStream_t stream) {
    const float* x        = (const float*)d_in[0];
    const int*   edge     = (const int*)  d_in[1];   // [2, E]
    const int*   batch    = (const int*)  d_in[2];
    const float* W_root   = (const float*)d_in[3];   // [L, D, D]
    const float* W_rel    = (const float*)d_in[4];   // [L, D, D]
    const float* b_conv   = (const float*)d_in[5];   // [L, D]
    const float* ln_gamma = (const float*)d_in[6];   // [L, D]
    const float* ln_beta  = (const float*)d_in[7];   // [L, D]
    const float* bW       = (const float*)d_in[8];   // [D, 1]
    const float* bb       = (const float*)d_in[9];   // [1]
    const float* dom_W1   = (const float*)d_in[10];  // [D, D/2]
    const float* dom_b1   = (const float*)d_in[11];  // [D/2]
    const float* dom_W2   = (const float*)d_in[12];  // [D/2, NPAT]
    const float* dom_b2   = (const float*)d_in[13];  // [NPAT]

    const int N = in_sizes[0] / D;       // 50000
    const int E = in_sizes[1] / 2;       // 600000
    const int L = in_sizes[3] / (D * D); // 4

    const int src_off = 0;               // edge_index[0]
    const int dst_off = E;               // edge_index[1]

    // Workspace layout (floats)
    float* Xc     = (float*)d_ws;                 // N*D
    float* H      = Xc + (size_t)N * D;           // N*D
    float* Y      = H  + (size_t)N * D;           // N*D
    float* pooled = Y  + (size_t)N * D;           // 64*128
    float* counts = pooled + NGRAPHS * D;         // 64

    float* out_boundary = (float*)d_out;          // [N]
    float* out_domain   = out_boundary + N;       // [64*50]

    // x -> mutable copy
    {
        int n = N * D;
        copy_kernel<<<(n + 255) / 256, 256, 0, stream>>>(x, Xc, n);
    }

    const int rowTiles   = N / 16;                        // 3125
    const int edgeBlocks = (E * 32 + 255) / 256;          // wave per edge
    const int nodeBlocks = (N * 32 + 255) / 256;          // wave per node

    for (int l = 0; l < L; ++l) {
        // H = Xc @ W_root[l] + b_conv[l];  Y = Xc @ W_rel[l]
        gemm2_wmma_kernel<<<rowTiles, 256, 0, stream>>>(
            Xc, W_root + (size_t)l * D * D, W_rel + (size_t)l * D * D,
            b_conv + (size_t)l * D, H, Y);
        // H[dst] += Y[src]
        edge_scatter_kernel<<<edgeBlocks, 256, 0, stream>>>(
            Y, edge + src_off, edge + dst_off, H, E);
        // Xc = elu(LN(H)) + Xc
        ln_elu_res_kernel<<<nodeBlocks, 256, 0, stream>>>(
            H, Xc, ln_gamma + (size_t)l * D, ln_beta + (size_t)l * D, N);
    }

    // boundary_logits
    boundary_kernel<<<nodeBlocks, 256, 0, stream>>>(Xc, bW, bb, out_boundary, N);

    // pooled mean per graph
    {
        int n = NGRAPHS * D + NGRAPHS;
        zero_kernel<<<(n + 255) / 256, 256, 0, stream>>>(pooled, n);
    }
    pool_kernel<<<nodeBlocks, 256, 0, stream>>>(Xc, batch, pooled, counts, N);

    // domain head
    domain_head_kernel<<<1, 256, 0, stream>>>(pooled, counts, dom_W1, dom_b1,
                                              dom_W2, dom_b2, out_domain);
}